// BasicBlock_73203422593428
// MI455X (gfx1250) — compile-verified
//
#include <hip/hip_runtime.h>
#include <hip/hip_bf16.h>

typedef __attribute__((ext_vector_type(2))) float v2f;
typedef __attribute__((ext_vector_type(8))) float v8f;

#define N_NODES 50000
#define N_EDGES 500000
#define D 128          // D_IN == D_OUT == 128

// ---------------------------------------------------------------------------
// 1) degree init: self-loop contributes 1 to every node
// ---------------------------------------------------------------------------
__global__ void deg_init_kernel(float* __restrict__ deg) {
    int i = blockIdx.x * blockDim.x + threadIdx.x;
    if (i < N_NODES) deg[i] = 1.0f;
}

// ---------------------------------------------------------------------------
// 2) degree accumulation over edge targets
// ---------------------------------------------------------------------------
__global__ void deg_accum_kernel(const long long* __restrict__ ei, float* __restrict__ deg) {
    int e = blockIdx.x * blockDim.x + threadIdx.x;
    if (e < N_EDGES) {
        int dst = (int)ei[N_EDGES + e];   // row 1 = aggregation target
        atomicAdd(deg + dst, 1.0f);
    }
}

// ---------------------------------------------------------------------------
// 3) dinv = rsqrt(deg) in place
// ---------------------------------------------------------------------------
__global__ void dinv_kernel(float* __restrict__ deg) {
    int i = blockIdx.x * blockDim.x + threadIdx.x;
    if (i < N_NODES) {
        float d = deg[i];
        deg[i] = (d > 0.0f) ? __frsqrt_rn(d) : 0.0f;
    }
}

// ---------------------------------------------------------------------------
// 4) xw = x @ W via V_WMMA_F32_16X16X4_F32 (full fp32 precision).
//    One wave -> one 16x16 output tile. Block = 256 threads = 8 waves
//    covering one 16-row stripe across all 128 output columns.
//    50000 % 16 == 0 -> no tails, EXEC stays all-ones (WMMA requirement).
// ---------------------------------------------------------------------------
__global__ void gemm_wmma_kernel(const float* __restrict__ x,
                                 const float* __restrict__ W,
                                 float* __restrict__ xw) {
    const int wave = threadIdx.x >> 5;         // 0..7 -> N tile
    const int lane = threadIdx.x & 31;
    const int r16  = lane & 15;
    const int hi   = lane >> 4;                // 0 or 1
    const int mBase = blockIdx.x * 16;
    const int nBase = wave * 16;

    // A: lane holds row M = mBase + r16, K-pair starting at 2*hi (+k)
    const float* arow = x + (size_t)(mBase + r16) * D + 2 * hi;
    // B: lane holds col N = nBase + r16, K-pair starting at 2*hi (+k)
    const float* bcol = W + nBase + r16;

    v8f c = {};
#pragma unroll
    for (int k = 0; k < D; k += 4) {
        v2f a, b;
        a.x = arow[k];
        a.y = arow[k + 1];
        const int kb = k + 2 * hi;
        b.x = bcol[(size_t)kb * D];
        b.y = bcol[(size_t)(kb + 1) * D];
        c = __builtin_amdgcn_wmma_f32_16x16x4_f32(
                /*neg_a=*/false, a, /*neg_b=*/false, b,
                /*c_mod=*/(short)0, c, /*reuse_a=*/false, /*reuse_b=*/false);
    }

    // D/C layout: VGPR r, lane l -> M = r + 8*(l>>4), N = l&15
#pragma unroll
    for (int r = 0; r < 8; ++r) {
        int m = mBase + r + 8 * hi;
        xw[(size_t)m * D + nBase + r16] = c[r];
    }
}

// ---------------------------------------------------------------------------
// 5) self-loop: out[i] = xw[i] * dinv[i]^2   (also initializes/de-poisons out)
//    One wave per node: 32 lanes x float4 = 128 channels.
// ---------------------------------------------------------------------------
__global__ void selfloop_kernel(const float* __restrict__ xw,
                                const float* __restrict__ dinv,
                                float* __restrict__ out) {
    int gid = blockIdx.x * blockDim.x + threadIdx.x;
    int i = gid >> 5;
    if (i >= N_NODES) return;
    int q = gid & 31;
    float di = dinv[i];
    float nrm = di * di;
    float4 v = ((const float4*)(xw + (size_t)i * D))[q];
    v.x *= nrm; v.y *= nrm; v.z *= nrm; v.w *= nrm;
    ((float4*)(out + (size_t)i * D))[q] = v;
}

// ---------------------------------------------------------------------------
// 6) edge scatter: out[dst] += xw[src] * dinv[src]*dinv[dst]
//    One wave per edge; float4 gather (coalesced 512B line, L2-resident) and
//    4 global_atomic_add_f32 per lane.
// ---------------------------------------------------------------------------
__global__ void scatter_kernel(const long long* __restrict__ ei,
                               const float* __restrict__ xw,
                               const float* __restrict__ dinv,
                               float* __restrict__ out) {
    int gid = blockIdx.x * blockDim.x + threadIdx.x;
    int e = gid >> 5;
    if (e >= N_EDGES) return;
    int q = gid & 31;
    int s = (int)ei[e];
    int d = (int)ei[N_EDGES + e];
    float nrm = dinv[s] * dinv[d];
    float4 v = ((const float4*)(xw + (size_t)s * D))[q];
    float* o = out + (size_t)d * D + q * 4;
    atomicAdd(o + 0, v.x * nrm);
    atomicAdd(o + 1, v.y * nrm);
    atomicAdd(o + 2, v.z * nrm);
    atomicAdd(o + 3, v.w * nrm);
}

// ---------------------------------------------------------------------------
// 7) out = relu(out + b) in place
// ---------------------------------------------------------------------------
__global__ void bias_relu_kernel(float* __restrict__ out, const float* __restrict__ b) {
    int gid = blockIdx.x * blockDim.x + threadIdx.x;
    if (gid >= N_NODES * (D / 4)) return;
    int q = gid & 31;                       // float4 index within row
    float4 v  = ((float4*)out)[gid];
    float4 bb = ((const float4*)b)[q];
    v.x = fmaxf(v.x + bb.x, 0.0f);
    v.y = fmaxf(v.y + bb.y, 0.0f);
    v.z = fmaxf(v.z + bb.z, 0.0f);
    v.w = fmaxf(v.w + bb.w, 0.0f);
    ((float4*)out)[gid] = v;
}

extern "C" void kernel_launch(void* const* d_in, const int* in_sizes, int n_in,
                              void* d_out, int out_size, void* d_ws, size_t ws_size,
                              hipStream_t stream) {
    const float*     x  = (const float*)d_in[0];
    const long long* ei = (const long long*)d_in[1];   // int64 [2, E]
    const float*     W  = (const float*)d_in[2];
    const float*     b  = (const float*)d_in[3];
    float*           out = (float*)d_out;

    // workspace: dinv (N floats, 200000 bytes, 16B-aligned) + xw (N*D floats)
    float* deg = (float*)d_ws;        // becomes dinv in place
    float* xw  = deg + N_NODES;

    deg_init_kernel <<<(N_NODES + 255) / 256, 256, 0, stream>>>(deg);
    deg_accum_kernel<<<(N_EDGES + 255) / 256, 256, 0, stream>>>(ei, deg);
    dinv_kernel     <<<(N_NODES + 255) / 256, 256, 0, stream>>>(deg);

    gemm_wmma_kernel<<<N_NODES / 16, 256, 0, stream>>>(x, W, xw);

    selfloop_kernel <<<(N_NODES * 32 + 255) / 256, 256, 0, stream>>>(xw, deg, out);
    scatter_kernel  <<<(N_EDGES * 32 + 255) / 256, 256, 0, stream>>>(ei, xw, deg, out);
    bias_relu_kernel<<<(N_NODES * 32 + 255) / 256, 256, 0, stream>>>(out, b);
}